// EMD_76055280878032
// MI455X (gfx1250) — compile-verified
//
#include <hip/hip_runtime.h>
#include <hip/hip_bf16.h>
#include <math.h>

// ---------------------------------------------------------------------------
// EMD + NTIL blend, MI455X (gfx1250, wave32).
// Memory-bound: reads x (1 GiB) exactly once via async global->LDS staging.
// ---------------------------------------------------------------------------

#define NTIL_LAMBDA 0.3f
#define EMD_EPS     1e-10f

#define C_CLS            4096
#define WAVES_PER_BLOCK  4
#define THREADS_PER_BLK  (WAVES_PER_BLOCK * 32)
#define ITERS_PER_LANE   (C_CLS / (32 * 4))   // 32 float4 iterations per lane

#if defined(__gfx1250__) && __has_builtin(__builtin_amdgcn_global_load_async_to_lds_b128)
#define HAVE_ASYNC_LDS 1
#else
#define HAVE_ASYNC_LDS 0
#endif

typedef int v4i __attribute__((vector_size(16)));

__device__ __forceinline__ void async_copy16(const float* g, float* l) {
#if HAVE_ASYNC_LDS
    __builtin_amdgcn_global_load_async_to_lds_b128(
        (__attribute__((address_space(1))) v4i*)g,
        (__attribute__((address_space(3))) v4i*)l,
        /*offset=*/0, /*cpol=*/0);
#else
    *reinterpret_cast<float4*>(l) = *reinterpret_cast<const float4*>(g);
#endif
}

__device__ __forceinline__ void wait_async_zero() {
#if HAVE_ASYNC_LDS
#if __has_builtin(__builtin_amdgcn_s_wait_asynccnt)
    __builtin_amdgcn_s_wait_asynccnt(0);
#else
    asm volatile("s_wait_asynccnt 0" ::: "memory");
#endif
#endif
    asm volatile("" ::: "memory");
}

__device__ __forceinline__ float wave_max32(float v) {
    #pragma unroll
    for (int o = 16; o > 0; o >>= 1) v = fmaxf(v, __shfl_xor(v, o, 32));
    return v;
}

__device__ __forceinline__ float wave_sum32(float v) {
    #pragma unroll
    for (int o = 16; o > 0; o >>= 1) v += __shfl_xor(v, o, 32);
    return v;
}

// Kernel 1: per-row softmax-EMD. One row per wave; row staged in LDS by the
// async DMA path so HBM traffic is exactly one pass over x.
__global__ __launch_bounds__(THREADS_PER_BLK)
void emd_rows_kernel(const float* __restrict__ x,
                     const int*   __restrict__ y,
                     float*       __restrict__ partials,
                     int nrows) {
    __shared__ float lds[WAVES_PER_BLOCK * C_CLS];   // 64 KB
    __shared__ float wpart[WAVES_PER_BLOCK];

    const int wave = threadIdx.x >> 5;
    const int lane = threadIdx.x & 31;
    const int row  = blockIdx.x * WAVES_PER_BLOCK + wave;

    float emd_row = 0.0f;
    if (row < nrows) {
        const float* grow = x + (size_t)row * C_CLS;
        float*       lrow = &lds[wave * C_CLS];

        // Stage 16KB row: 32 async b128 ops x 32 lanes x 16B. ASYNCcnt-tracked,
        // per-wave, so no workgroup barrier is needed before consuming it.
        #pragma unroll 8
        for (int it = 0; it < ITERS_PER_LANE; ++it) {
            const int off = (it * 32 + lane) * 4;   // float index
            async_copy16(grow + off, lrow + off);
        }
        wait_async_zero();

        // Pass 1: row max (stable softmax).
        float m = -INFINITY;
        #pragma unroll 8
        for (int it = 0; it < ITERS_PER_LANE; ++it) {
            const float4 v = *reinterpret_cast<const float4*>(&lrow[(it * 32 + lane) * 4]);
            m = fmaxf(m, fmaxf(fmaxf(v.x, v.y), fmaxf(v.z, v.w)));
        }
        m = wave_max32(m);

        // Pass 2: l = sum(exp), s = sum(|y - j| * exp). One v_exp_f32 per element.
        const float yf = (float)y[row];
        float l = 0.0f, s = 0.0f;
        #pragma unroll 8
        for (int it = 0; it < ITERS_PER_LANE; ++it) {
            const int   j0 = (it * 32 + lane) * 4;
            const float4 v = *reinterpret_cast<const float4*>(&lrow[j0]);
            const float e0 = __expf(v.x - m);
            const float e1 = __expf(v.y - m);
            const float e2 = __expf(v.z - m);
            const float e3 = __expf(v.w - m);
            l += (e0 + e1) + (e2 + e3);
            s += fabsf(yf - (float)(j0    )) * e0
               + fabsf(yf - (float)(j0 + 1)) * e1
               + fabsf(yf - (float)(j0 + 2)) * e2
               + fabsf(yf - (float)(j0 + 3)) * e3;
        }
        l = wave_sum32(l);
        s = wave_sum32(s);
        emd_row = s / l;
    }

    if (lane == 0) wpart[wave] = emd_row;
    __syncthreads();
    if (threadIdx.x == 0) {
        float acc = 0.0f;
        #pragma unroll
        for (int w = 0; w < WAVES_PER_BLOCK; ++w) acc += wpart[w];
        partials[blockIdx.x] = acc;     // deterministic: one writer per slot
    }
}

// Kernel 2: deterministic fixed-order reduction of block partials + NTIL blend.
__global__ __launch_bounds__(256)
void emd_finalize_kernel(const float* __restrict__ partials, int npart,
                         const float* __restrict__ mle_ptr,
                         float*       __restrict__ out, int nrows) {
    __shared__ double sdata[256];
    const int t = threadIdx.x;
    double acc = 0.0;
    for (int i = t; i < npart; i += 256) acc += (double)partials[i];
    sdata[t] = acc;
    __syncthreads();
    #pragma unroll
    for (int sft = 128; sft > 0; sft >>= 1) {
        if (t < sft) sdata[t] += sdata[t + sft];
        __syncthreads();
    }
    if (t == 0) {
        const float emd = (float)(sdata[0] / (double)nrows);
        const float mle = *mle_ptr;
        // stop_gradient(mle/(emd+eps)) * lambda * emd + mle  (forward value)
        const float norm_emd = (mle / (emd + EMD_EPS)) * NTIL_LAMBDA * emd;
        out[0] = norm_emd + mle;
    }
}

extern "C" void kernel_launch(void* const* d_in, const int* in_sizes, int n_in,
                              void* d_out, int out_size, void* d_ws, size_t ws_size,
                              hipStream_t stream) {
    const float* x   = (const float*)d_in[0];
    const int*   y   = (const int*)d_in[1];
    const float* mle = (const float*)d_in[2];
    float*       out = (float*)d_out;

    const int nrows  = in_sizes[1];                                  // N
    const int blocks = (nrows + WAVES_PER_BLOCK - 1) / WAVES_PER_BLOCK;

    float* partials = (float*)d_ws;                                  // blocks floats

    emd_rows_kernel<<<blocks, THREADS_PER_BLK, 0, stream>>>(x, y, partials, nrows);
    emd_finalize_kernel<<<1, 256, 0, stream>>>(partials, blocks, mle, out, nrows);
}